// Word2VecModel_77446850281559
// MI455X (gfx1250) — compile-verified
//
#include <hip/hip_runtime.h>

// f32 WMMA types: A/B = 16x4 / 4x16 f32 -> 2 VGPRs per lane; C/D = 16x16 f32 -> 8 VGPRs.
typedef __attribute__((ext_vector_type(2))) float v2f;
typedef __attribute__((ext_vector_type(8))) float v8f;

#define W2V_H 128
#define W2V_N 5

__device__ __forceinline__ float softplus_f(float x) {
  // numerically stable: max(x,0) + log1p(exp(-|x|))
  float ax = __builtin_fabsf(x);
  return fmaxf(x, 0.0f) + log1pf(__expf(-ax));
}

// One wave32 handles 16 batch rows and all 6 targets (1 true + 5 sampled).
// D = A(16x4) * B(4x16) accumulated over K=128 in chunks of 4; diagonal D[m,m]
// is the dot product for batch row m against its own target column m.
__global__ __launch_bounds__(256) void w2v_nce_wmma_kernel(
    const int* __restrict__ inputs,   // [B]
    const int* __restrict__ labels,   // [B]
    const int* __restrict__ sampled,  // [N, B]
    const float* __restrict__ syn0,   // [V, H]
    const float* __restrict__ syn1,   // [V, H]
    float* __restrict__ out,          // [B, 1+N]
    int Bn) {
  const int lane  = threadIdx.x & 31;
  const int group = (blockIdx.x * blockDim.x + threadIdx.x) >> 5;  // wave id
  const int m16   = lane & 15;   // batch row within group (A row / B col)
  const int half  = lane >> 4;   // K-pair selector within a K=4 chunk
  const int row   = group * 16 + m16;

  // Gather indices for this row (lanes 16..31 redundantly load the same).
  const int idx_in = inputs[row];
  int tgt[6];
  tgt[0] = labels[row];
#pragma unroll
  for (int j = 0; j < W2V_N; ++j) tgt[1 + j] = sampled[j * Bn + row];

  // Per-lane base pointers matching the WMMA f32 A/B VGPR layouts:
  // lane L supplies elements k = K0 + 2*(L/16) and k+1 (an 8-byte load).
  const float* aPtr = syn0 + (size_t)idx_in * W2V_H + 2 * half;
  const float* bPtr[6];
#pragma unroll
  for (int j = 0; j < 6; ++j) bPtr[j] = syn1 + (size_t)tgt[j] * W2V_H + 2 * half;

  v8f acc[6];
#pragma unroll
  for (int j = 0; j < 6; ++j) acc[j] = (v8f)0.0f;

  // K loop: 32 chunks of 4. Uniform control flow -> EXEC stays all-ones for WMMA.
  for (int k0 = 0; k0 < W2V_H; k0 += 4) {
    v2f a = *(const v2f*)(aPtr + k0);
#pragma unroll
    for (int j = 0; j < 6; ++j) {
      v2f b = *(const v2f*)(bPtr[j] + k0);
      // 8 args: (neg_a, A, neg_b, B, c_mod, C, reuse_a, reuse_b)
      acc[j] = __builtin_amdgcn_wmma_f32_16x16x4_f32(
          false, a, false, b, (short)0, acc[j], false, false);
    }
  }

  // Diagonal extraction (C/D layout: VGPR r, lanes 0-15 -> D[r, lane];
  // lanes 16-31 -> D[8+r, lane-16]). Diagonal lives in lanes 0-7 and 24-31.
  const bool diag = (lane < 8) || (lane >= 24);
  if (diag) {
    const int r = lane & 7;                       // accumulator VGPR index
    const int m = (lane < 8) ? lane : (lane - 16);  // batch row within group
    float* o = out + (size_t)(group * 16 + m) * (1 + W2V_N);
    const float tl = acc[0][r];
    o[0] = softplus_f(-tl);  // true label: BCE target=1 -> softplus(-x)
#pragma unroll
    for (int j = 1; j < 6; ++j) o[j] = softplus_f(acc[j][r]);  // target=0 -> softplus(x)
  }
}

extern "C" void kernel_launch(void* const* d_in, const int* in_sizes, int n_in,
                              void* d_out, int out_size, void* d_ws, size_t ws_size,
                              hipStream_t stream) {
  const int*   inputs  = (const int*)d_in[0];
  const int*   labels  = (const int*)d_in[1];
  const int*   sampled = (const int*)d_in[2];
  const float* syn0    = (const float*)d_in[3];
  const float* syn1    = (const float*)d_in[4];
  float*       out     = (float*)d_out;

  const int Bn = in_sizes[0];          // 16384
  const int groups = Bn / 16;          // 1024 waves of work
  const int block = 256;               // 8 waves per block (wave32)
  const int wavesPerBlock = block / 32;
  const int grid = (groups + wavesPerBlock - 1) / wavesPerBlock;

  w2v_nce_wmma_kernel<<<grid, block, 0, stream>>>(inputs, labels, sampled,
                                                  syn0, syn1, out, Bn);
}